// MultimodalAtt_34763465294204
// MI455X (gfx1250) — compile-verified
//
#include <hip/hip_runtime.h>
#include <math.h>

// ---------------------------------------------------------------------------
// MI455X (gfx1250) implementation.
//  * All GEMMs go through one wave32 WMMA kernel: each wave owns a 16x16 f32
//    output tile and marches K with V_WMMA_F32_16X16X4_F32 (full fp32 matrix
//    path -> matches reference numerics), unrolled 4x (K step 16).
//  * A-fragment (16x4 f32): lane = m%16 + 16*(k/2), vgpr = k%2  (ISA 7.12.2)
//  * B-fragment (4x16 f32): mirror with n = lane%16 (sparse-B table pattern)
//  * C/D (16x16 f32): vgpr i -> M = i + 8*(lane/16), N = lane%16
//  * Loads are UNCONDITIONAL float2 (global_load_b64) from row-clamped
//    pointers: every call site has M%16==0 and K%4==0, and out-of-range N
//    lanes are masked only in the epilogue. Keeps the K-loop branch-free.
//  * Sequential parts (LSTM recurrences, decoder) are host-ordered launches;
//    all weights (~25 MB) stay resident in the 192 MB L2 across 29 steps.
// ---------------------------------------------------------------------------

typedef __attribute__((ext_vector_type(2)))  float     v2f;
typedef __attribute__((ext_vector_type(8)))  float     v8f;

#define GF_ACC  1   // D = prev(D) + A@W^T (+bias)
#define GF_TANH 2   // D = tanh(result)

__device__ __forceinline__ float sigf(float x) { return 1.0f / (1.0f + expf(-x)); }

// D[M,N] (ldd) = [acc] A[M,K](lda) @ W[N,K](ldw)^T + bias1 + bias2, opt tanh.
// Requires: M % 16 == 0, K % 4 == 0, all row pointers 8-byte aligned.
__global__ __launch_bounds__(256)
void gemm_wmma(const float* __restrict__ A, int lda,
               const float* __restrict__ W, int ldw,
               const float* __restrict__ bias1, const float* __restrict__ bias2,
               float* __restrict__ D, int ldd,
               int M, int N, int K, int flags)
{
  const int wave  = threadIdx.x >> 5;       // 8 waves: 2 (M) x 4 (N)
  const int lane  = threadIdx.x & 31;
  const int tileM = (blockIdx.x * 2 + (wave >> 2)) * 16;
  const int tileN = (blockIdx.y * 4 + (wave & 3)) * 16;
  if (tileM >= M || tileN >= N) return;     // wave-uniform: EXEC stays all-1s
  const int  half  = lane >> 4;
  const int  lr    = lane & 15;
  const int  row   = tileM + lr;            // A row for this lane (always < M)
  const int  col   = tileN + lr;            // W row (= output column)
  const bool colOK = col < N;
  // Clamp to a valid row so loads never fault; bad lanes masked in epilogue.
  const float* Ap = A + (size_t)row * lda + half * 2;
  const float* Wp = W + (size_t)(colOK ? col : (N - 1)) * ldw + half * 2;

  v8f acc = {0.f,0.f,0.f,0.f,0.f,0.f,0.f,0.f};

  int k0 = 0;
  const int Kmain = K & ~15;
  for (; k0 < Kmain; k0 += 16) {            // 4 WMMAs, 8 b64 loads per iter
    v2f a0 = *(const v2f*)(Ap + k0);
    v2f b0 = *(const v2f*)(Wp + k0);
    v2f a1 = *(const v2f*)(Ap + k0 + 4);
    v2f b1 = *(const v2f*)(Wp + k0 + 4);
    v2f a2 = *(const v2f*)(Ap + k0 + 8);
    v2f b2 = *(const v2f*)(Wp + k0 + 8);
    v2f a3 = *(const v2f*)(Ap + k0 + 12);
    v2f b3 = *(const v2f*)(Wp + k0 + 12);
    acc = __builtin_amdgcn_wmma_f32_16x16x4_f32(false, a0, false, b0, (short)0, acc, false, false);
    acc = __builtin_amdgcn_wmma_f32_16x16x4_f32(false, a1, false, b1, (short)0, acc, false, false);
    acc = __builtin_amdgcn_wmma_f32_16x16x4_f32(false, a2, false, b2, (short)0, acc, false, false);
    acc = __builtin_amdgcn_wmma_f32_16x16x4_f32(false, a3, false, b3, (short)0, acc, false, false);
  }
  for (; k0 < K; k0 += 4) {                 // K remainder (K%4 == 0 always)
    v2f a = *(const v2f*)(Ap + k0);
    v2f b = *(const v2f*)(Wp + k0);
    acc = __builtin_amdgcn_wmma_f32_16x16x4_f32(false, a, false, b, (short)0, acc, false, false);
  }

  if (!colOK) return;
  float bv = 0.f;
  if (bias1) bv += bias1[col];
  if (bias2) bv += bias2[col];
#pragma unroll
  for (int i = 0; i < 8; ++i) {
    int m = tileM + i + 8 * half;           // m always < M (M % 16 == 0)
    size_t o = (size_t)m * ldd + col;
    float val = acc[i] + bv;
    if (flags & GF_ACC)  val += D[o];
    if (flags & GF_TANH) val = tanhf(val);
    D[o] = val;
  }
}

__global__ void fill_kernel(float* p, float v, int n) {
  int i = blockIdx.x * blockDim.x + threadIdx.x;
  if (i < n) p[i] = v;
}

// out[b*L+s, :] = emb[idx[b*Lsrc+s], :]
__global__ void gather_kernel(const float* __restrict__ emb, const int* __restrict__ idx,
                              float* __restrict__ out, int B_, int L, int Lsrc, int DW_)
{
  size_t i = (size_t)blockIdx.x * blockDim.x + threadIdx.x;
  size_t total = (size_t)B_ * L * DW_;
  if (i >= total) return;
  int d = (int)(i % DW_);
  size_t r = i / DW_;
  int s = (int)(r % L);
  int b = (int)(r / L);
  int tok = idx[b * Lsrc + s];
  out[i] = emb[(size_t)tok * DW_ + d];
}

// ac4s[b,t,d] = sum_{g<16} ac4[b,t,g,d] + fc2h[b*512+d]   (T=20, D=512)
__global__ void ac4sum_kernel(const float* __restrict__ ac4, const float* __restrict__ fc2h,
                              float* __restrict__ out, int total)
{
  int i = blockIdx.x * blockDim.x + threadIdx.x;
  if (i >= total) return;
  int d  = i % 512;
  int bt = i / 512;
  int b  = bt / 20;
  const float* p = ac4 + (size_t)bt * 16 * 512 + d;
  float s = 0.f;
#pragma unroll
  for (int g = 0; g < 16; ++g) s += p[g * 512];
  out[i] = s + fc2h[b * 512 + d];
}

// LSTM cell: g = g1[b,ls1] (+ g2[b,ls2]); split i,f,gg,o at H; update h,c; opt write h.
__global__ void lstm_cell_kernel(const float* __restrict__ g1, int ls1,
                                 const float* __restrict__ g2, int ls2,
                                 float* __restrict__ h, float* __restrict__ c,
                                 float* __restrict__ yout, int ldy, int H_, int B_)
{
  int i = blockIdx.x * blockDim.x + threadIdx.x;
  if (i >= B_ * H_) return;
  int b = i / H_, j = i % H_;
  size_t r1 = (size_t)b * ls1;
  float gi = g1[r1 + j], gf = g1[r1 + H_ + j], gg = g1[r1 + 2*H_ + j], go = g1[r1 + 3*H_ + j];
  if (g2) {
    size_t r2 = (size_t)b * ls2;
    gi += g2[r2 + j]; gf += g2[r2 + H_ + j]; gg += g2[r2 + 2*H_ + j]; go += g2[r2 + 3*H_ + j];
  }
  float cc = sigf(gf) * c[i] + sigf(gi) * tanhf(gg);
  float hh = sigf(go) * tanhf(cc);
  c[i] = cc; h[i] = hh;
  if (yout) yout[(size_t)b * ldy + j] = hh;
}

// scores[s] = sum_e tanh(pre[b,s,e]+hterm[b,e])*v[e]; a = softmax_s; out[b,:] = sum_s a*src[b,s,:]
__global__ __launch_bounds__(256)
void attpool_kernel(const float* __restrict__ pre, const float* __restrict__ hterm,
                    const float* __restrict__ v, const float* __restrict__ src,
                    float* __restrict__ out, int S, int E, int Dv)
{
  int b = blockIdx.x, tid = threadIdx.x;
  __shared__ float red[256];
  __shared__ float sc[32];
  __shared__ float aw[32];
  const float* hp = hterm + (size_t)b * E;
  for (int s = 0; s < S; ++s) {
    const float* pp = pre + ((size_t)b * S + s) * E;
    float p = 0.f;
    for (int e = tid; e < E; e += 256) p += tanhf(pp[e] + hp[e]) * v[e];
    red[tid] = p; __syncthreads();
    for (int w = 128; w > 0; w >>= 1) { if (tid < w) red[tid] += red[tid + w]; __syncthreads(); }
    if (tid == 0) sc[s] = red[0];
    __syncthreads();
  }
  if (tid < S) {
    float m = -3.4e38f;
    for (int s = 0; s < S; ++s) m = fmaxf(m, sc[s]);
    float den = 0.f;
    for (int s = 0; s < S; ++s) den += expf(sc[s] - m);
    aw[tid] = expf(sc[tid] - m) / den;
  }
  __syncthreads();
  for (int d = tid; d < Dv; d += 256) {
    float acc = 0.f;
    for (int s = 0; s < S; ++s) acc += aw[s] * src[((size_t)b * S + s) * Dv + d];
    out[(size_t)b * Dv + d] = acc;
  }
}

// BiDAF-style flow: Hm (B,20,512), U (B,30,512) -> G (B,20,2048). One block per batch.
__global__ __launch_bounds__(256)
void flow_kernel(const float* __restrict__ Hm, const float* __restrict__ U,
                 const float* __restrict__ w1, const float* __restrict__ w2,
                 const float* __restrict__ w3, float* __restrict__ G)
{
  const int TT = 20, SS = 30, D = 512;
  int b = blockIdx.x, tid = threadIdx.x;
  __shared__ float S[20][30];
  __shared__ float hw1[20], uw2[30], rowm[20], bw[20], ht[512];

  if (tid < TT) {
    const float* hp = Hm + ((size_t)b * TT + tid) * D;
    float s = 0.f; for (int d = 0; d < D; ++d) s += hp[d] * w1[d];
    hw1[tid] = s;
  } else if (tid >= 32 && tid < 32 + SS) {
    int j = tid - 32;
    const float* up = U + ((size_t)b * SS + j) * D;
    float s = 0.f; for (int d = 0; d < D; ++d) s += up[d] * w2[d];
    uw2[j] = s;
  }
  __syncthreads();
  for (int p = tid; p < TT * SS; p += blockDim.x) {
    int i = p / SS, j = p % SS;
    const float* hp = Hm + ((size_t)b * TT + i) * D;
    const float* up = U + ((size_t)b * SS + j) * D;
    float s = 0.f; for (int d = 0; d < D; ++d) s += hp[d] * w3[d] * up[d];
    S[i][j] = s + hw1[i] + uw2[j];
  }
  __syncthreads();
  if (tid < TT) {                       // row softmax over j, keep row max
    float m = -3.4e38f;
    for (int j = 0; j < SS; ++j) m = fmaxf(m, S[tid][j]);
    rowm[tid] = m;
    float den = 0.f;
    for (int j = 0; j < SS; ++j) den += expf(S[tid][j] - m);
    for (int j = 0; j < SS; ++j) S[tid][j] = expf(S[tid][j] - m) / den;
  }
  __syncthreads();
  if (tid < TT) {                       // bw = softmax_i(rowmax)
    float m = -3.4e38f;
    for (int i = 0; i < TT; ++i) m = fmaxf(m, rowm[i]);
    float den = 0.f;
    for (int i = 0; i < TT; ++i) den += expf(rowm[i] - m);
    bw[tid] = expf(rowm[tid] - m) / den;
  }
  __syncthreads();
  for (int d = tid; d < D; d += blockDim.x) {
    float s = 0.f;
    for (int i = 0; i < TT; ++i) s += bw[i] * Hm[((size_t)b * TT + i) * D + d];
    ht[d] = s;
  }
  __syncthreads();
  for (int i = 0; i < TT; ++i) {
    float* g = G + ((size_t)b * TT + i) * 4 * D;
    for (int d = tid; d < D; d += blockDim.x) {
      float hm = Hm[((size_t)b * TT + i) * D + d];
      float ut = 0.f;
      for (int j = 0; j < SS; ++j) ut += S[i][j] * U[((size_t)b * SS + j) * D + d];
      g[d]         = hm;
      g[D + d]     = ut;
      g[2 * D + d] = hm * ut;
      g[3 * D + d] = hm * ht[d];
    }
  }
}

// ---------------------------------------------------------------------------

extern "C" void kernel_launch(void* const* d_in, const int* in_sizes, int n_in,
                              void* d_out, int out_size, void* d_ws, size_t ws_size,
                              hipStream_t stream)
{
  (void)in_sizes; (void)n_in; (void)out_size; (void)ws_size;
  const int B_ = 128, T_ = 20, ML_ = 30, Hh = 256, D2_ = 512, DW_ = 300, VOC = 4506;

  auto F = [&](int i) { return (const float*)d_in[i]; };
  auto Ii = [&](int i) { return (const int*)d_in[i]; };

  const float* ac4  = F(0);
  const float* afc2 = F(1);
  const int*   sem  = Ii(2);
  const int*   tgt  = Ii(3);
  const float* emb  = F(4);
  // fc2_p: 5..12  c4_p: 13..20  we_p: 21..28  (each: Wf,Uf,bif,bhf,Wb,Ub,bib,bhb)
  const float* decWih = F(29); const float* decWhh = F(30);
  const float* decbih = F(31); const float* decbhh = F(32);
  const float* attWh = F(33); const float* attWe = F(34);
  const float* attb  = F(35); const float* attv  = F(36);
  const float* fw1 = F(37); const float* fw2 = F(38); const float* fw3 = F(39);
  const float* tafWh = F(40); const float* tafWa = F(41);
  const float* tafb  = F(42); const float* tafv  = F(43);
  const float* WdW = F(44); const float* Wdb = F(45);
  const float* WoW = F(46); const float* Wob = F(47);

  float* ws = (float*)d_ws;
  size_t off = 0;
  auto alloc = [&](size_t n) { float* p = ws + off; off += n; return p; };
  float* SEMEMB = alloc((size_t)B_ * ML_ * DW_);        // (B,30,300)
  float* WTSEMB = alloc((size_t)B_ * (ML_-1) * DW_);    // (B,29,300)
  float* XG     = alloc((size_t)B_ * ML_ * 1024);       // per-direction xg
  float* WOUT   = alloc((size_t)B_ * ML_ * 512);        // w_out (B,30,512)
  float* HTMP   = alloc((size_t)2 * B_ * Hh);
  float* CTMP   = alloc((size_t)2 * B_ * Hh);
  float* FC2H   = alloc((size_t)2 * B_ * Hh);           // [hf;hb] flat (B,512) view
  float* AC4S   = alloc((size_t)B_ * T_ * D2_);
  float* C4OUT  = alloc((size_t)B_ * T_ * D2_);
  float* DECH   = alloc((size_t)B_ * D2_);              // [hf;hb] = decoder h (B,512)
  float* DECC   = alloc((size_t)B_ * D2_);
  float* GATES  = alloc((size_t)B_ * 2048);
  float* ENCWE  = alloc((size_t)B_ * T_ * D2_);
  float* HWH    = alloc((size_t)B_ * D2_);
  float* GBUF   = alloc((size_t)B_ * T_ * 2048);
  float* GWA    = alloc((size_t)B_ * T_ * D2_);         // G@Wa^T + b (precomputed)
  float* TAFOUT = alloc((size_t)B_ * 2048);
  float* DBUF0  = alloc((size_t)B_ * D2_);
  float* DBUF1  = alloc((size_t)B_ * D2_);

  auto gemm = [&](const float* A, int lda, const float* W, int ldw,
                  const float* b1, const float* b2,
                  float* D, int ldd, int M, int N, int K, int flags) {
    dim3 g((unsigned)((M + 31) / 32), (unsigned)((N + 63) / 64));
    gemm_wmma<<<g, 256, 0, stream>>>(A, lda, W, ldw, b1, b2, D, ldd, M, N, K, flags);
  };
  auto fill0 = [&](float* p, int n) {
    fill_kernel<<<(n + 255) / 256, 256, 0, stream>>>(p, 0.f, n);
  };
  auto cell = [&](const float* g1, int ls1, const float* g2, int ls2,
                  float* h, float* c, float* yout, int ldy, int Hc) {
    int n = B_ * Hc;
    lstm_cell_kernel<<<(n + 255) / 256, 256, 0, stream>>>(g1, ls1, g2, ls2, h, c, yout, ldy, Hc, B_);
  };
  // BiLSTM: x (B,T,Din) -> ys (B,T,512) optional; final [hf;hb] into hbuf/cbuf.
  auto bilstm = [&](const float* x, int T, int Din, int wbase,
                    float* ys, float* hbuf, float* cbuf) {
    for (int dir = 0; dir < 2; ++dir) {
      const float* Wih = F(wbase + dir * 4 + 0);
      const float* Whh = F(wbase + dir * 4 + 1);
      const float* bi  = F(wbase + dir * 4 + 2);
      const float* bh  = F(wbase + dir * 4 + 3);
      float* h = hbuf + (size_t)dir * B_ * Hh;
      float* c = cbuf + (size_t)dir * B_ * Hh;
      gemm(x, Din, Wih, Din, bi, bh, XG, 1024, B_ * T, 1024, Din, 0);   // xg = x@Wih^T+bi+bh
      fill0(h, B_ * Hh);
      fill0(c, B_ * Hh);
      for (int s = 0; s < T; ++s) {
        int t = dir ? (T - 1 - s) : s;
        gemm(h, Hh, Whh, Hh, nullptr, nullptr, GATES, 1024, B_, 1024, Hh, 0);
        float* yo = ys ? (ys + (size_t)t * 512 + dir * Hh) : nullptr;
        cell(GATES, 1024, XG + (size_t)t * 1024, T * 1024, h, c, yo, T * 512, Hh);
      }
    }
  };

  // 1) embedding gathers
  {
    int n1 = B_ * ML_ * DW_;
    gather_kernel<<<(n1 + 255) / 256, 256, 0, stream>>>(emb, sem, SEMEMB, B_, ML_, ML_, DW_);
    int n2 = B_ * (ML_ - 1) * DW_;
    gather_kernel<<<(n2 + 255) / 256, 256, 0, stream>>>(emb, tgt, WTSEMB, B_, ML_ - 1, ML_, DW_);
  }
  // 2) word-encoder BiLSTM -> WOUT (B,30,512)
  bilstm(SEMEMB, ML_, DW_, 21, WOUT, HTMP, CTMP);
  // 3) fc2 BiLSTM -> final hidden [hf;hb] (flat (B,512) view, matches reshape)
  bilstm(afc2, T_, 128, 5, nullptr, FC2H, CTMP);
  // 4) ac4s = sum over 4x4 grid + fc2_h broadcast
  { int n = B_ * T_ * D2_;
    ac4sum_kernel<<<(n + 255) / 256, 256, 0, stream>>>(ac4, FC2H, AC4S, n); }
  // 5) c4 BiLSTM -> C4OUT, dec_h/dec_c as flat [hf;hb] views
  bilstm(AC4S, T_, D2_, 13, C4OUT, DECH, DECC);
  // 6) dec_in = _att(dec_h, c4_out) -> DBUF0
  gemm(C4OUT, 512, attWe, 512, attb, nullptr, ENCWE, 512, B_ * T_, 512, 512, 0);
  gemm(DECH, 512, attWh, 512, nullptr, nullptr, HWH, 512, B_, 512, 512, 0);
  attpool_kernel<<<B_, 256, 0, stream>>>(ENCWE, HWH, attv, C4OUT, DBUF0, T_, 512, 512);
  // 7) G = _flow(c4_out, w_out)
  flow_kernel<<<B_, 256, 0, stream>>>(C4OUT, WOUT, fw1, fw2, fw3, GBUF);
  // 8) precompute G@Wa^T + b once (reference recomputes it every step — hoisted)
  gemm(GBUF, 2048, tafWa, 2048, tafb, nullptr, GWA, 512, B_ * T_, 512, 2048, 0);
  // 9) decoder: 29 steps
  for (int t = 0; t < ML_ - 1; ++t) {
    float* di = (t & 1) ? DBUF1 : DBUF0;
    float* ot = (t & 1) ? DBUF0 : DBUF1;
    // gates = [di|wt]@Wih^T + bih + bhh + h@Whh^T   (concat split by Wih columns)
    gemm(di, 512, decWih, 812, decbih, decbhh, GATES, 2048, B_, 2048, 512, 0);
    gemm(WTSEMB + (size_t)t * DW_, (ML_ - 1) * DW_, decWih + 512, 812,
         nullptr, nullptr, GATES, 2048, B_, 2048, DW_, GF_ACC);
    gemm(DECH, 512, decWhh, 512, nullptr, nullptr, GATES, 2048, B_, 2048, 512, GF_ACC);
    cell(GATES, 2048, nullptr, 0, DECH, DECC, nullptr, 0, 512);
    // taf attention over G
    gemm(DECH, 512, tafWh, 512, nullptr, nullptr, HWH, 512, B_, 512, 512, 0);
    attpool_kernel<<<B_, 256, 0, stream>>>(GWA, HWH, tafv, GBUF, TAFOUT, T_, 512, 2048);
    // out = tanh([taf|h]@WdW^T + Wdb)  (column-split, accumulate, tanh on 2nd)
    gemm(TAFOUT, 2048, WdW, 2560, Wdb, nullptr, ot, 512, B_, 512, 2048, 0);
    gemm(DECH, 512, WdW + 2048, 2560, nullptr, nullptr, ot, 512, B_, 512, 512, GF_ACC | GF_TANH);
    // logits -> d_out[b, t, :]
    gemm(ot, 512, WoW, 512, Wob, nullptr,
         (float*)d_out + (size_t)t * VOC, (ML_ - 1) * VOC, B_, VOC, 512, 0);
  }
}